// RPE_NKA_53678501265929
// MI455X (gfx1250) — compile-verified
//
#include <hip/hip_runtime.h>
#include <hip/hip_bf16.h>
#include <math.h>

typedef float v2f __attribute__((ext_vector_type(2)));
typedef float v8f __attribute__((ext_vector_type(8)));

#define Hh    8
#define Nn    1024
#define Dd    64
#define PLANE (Hh*Nn*Dd)          // 524288 floats
#define FLEN  1032                // DFT length = N + n_var(=H)=8 pad rows
#define WS_PART   0               // 256 partial sums
#define WS_NORM   256             // 1 float: 1/||k||_F
#define WS_PLANES 512             // 4 planes of PLANE floats: gre,gim,kre,kim

// ---------------- pass 1a: partial sums of k^2 ----------------
__global__ void rpe_sumsq(const float* __restrict__ k, float* __restrict__ ws) {
  __shared__ float red[256];
  int tid = threadIdx.x;
  float s = 0.f;
  for (int i = blockIdx.x * 256 + tid; i < PLANE; i += 256 * 256) {
    float x = k[i];
    s += x * x;
  }
  red[tid] = s;
  __syncthreads();
  for (int off = 128; off > 0; off >>= 1) {
    if (tid < off) red[tid] += red[tid + off];
    __syncthreads();
  }
  if (tid == 0) ws[WS_PART + blockIdx.x] = red[0];
}

// ---------------- pass 1b: finalize 1/||k|| ----------------
__global__ void rpe_finnorm(float* __restrict__ ws) {
  __shared__ float red[256];
  int tid = threadIdx.x;
  red[tid] = ws[WS_PART + tid];
  __syncthreads();
  for (int off = 128; off > 0; off >>= 1) {
    if (tid < off) red[tid] += red[tid + off];
    __syncthreads();
  }
  if (tid == 0) ws[WS_NORM] = 1.0f / sqrtf(red[0]);
}

// ---------------- pass 2: elu, s_k, 8-point head DFT -> 4 real planes ------
// block = one sequence row r (64 threads, one per feature z)
__global__ void rpe_hdft(const float* __restrict__ kin,
                         const float* __restrict__ vin,
                         float* __restrict__ ws) {
  __shared__ float red[8][64];
  __shared__ float sk[8];
  const int r = blockIdx.x;
  const int z = threadIdx.x;
  const float invn = ws[WS_NORM];

  float ke[8], vv[8];
#pragma unroll
  for (int hp = 0; hp < 8; ++hp) {
    int idx = (hp * Nn + r) * Dd + z;
    float x = kin[idx] * invn;
    ke[hp] = (x > 0.f) ? x : (expf(x) - 1.f);   // elu
    vv[hp] = vin[idx];
    red[hp][z] = ke[hp];
  }
  __syncthreads();
  if (z < 8) {                                   // s_k[h'] = sum_z k_[h',r,z]
    float s = 0.f;
#pragma unroll 8
    for (int j = 0; j < 64; ++j) s += red[z][j];
    sk[z] = s;
  }
  __syncthreads();

  float g[8];
#pragma unroll
  for (int hp = 0; hp < 8; ++hp) g[hp] = sk[hp] * vv[hp];

  // 8-point DFT across heads: u_hat[h] = sum_{h'} x[h'] * exp(-2*pi*i*h*h'/8)
  const float C8[8] = { 1.f,  0.70710678118654752f, 0.f, -0.70710678118654752f,
                       -1.f, -0.70710678118654752f, 0.f,  0.70710678118654752f};
  const float S8[8] = { 0.f,  0.70710678118654752f, 1.f,  0.70710678118654752f,
                        0.f, -0.70710678118654752f,-1.f, -0.70710678118654752f};
  float* gre = ws + WS_PLANES;
  float* gim = gre + PLANE;
  float* kre = gim + PLANE;
  float* kim = kre + PLANE;
#pragma unroll
  for (int h = 0; h < 8; ++h) {
    float gRe = 0.f, gIm = 0.f, kRe = 0.f, kIm = 0.f;
#pragma unroll
    for (int hp = 0; hp < 8; ++hp) {
      int t = (h * hp) & 7;
      gRe += C8[t] * g[hp];  gIm -= S8[t] * g[hp];
      kRe += C8[t] * ke[hp]; kIm -= S8[t] * ke[hp];
    }
    int o = (h * Nn + r) * Dd + z;
    gre[o] = gRe; gim[o] = gIm; kre[o] = kRe; kim[o] = kIm;
  }
}

// ---------------- pass 3: 1032-point row DFT as fp32 WMMA GEMM -------------
static __device__ __forceinline__ v8f wmma4(v2f a, v2f b, v8f c) {
  // D = A(16x4 f32) x B(4x16 f32) + C(16x16 f32)
  return __builtin_amdgcn_wmma_f32_16x16x4_f32(false, a, false, b,
                                               (short)0, c, false, false);
}

__global__ void rpe_wmma_dft(const float* __restrict__ ws,
                             float2* __restrict__ out) {
  // one wave = one 16x16 output tile of one head; 8 heads * 64 * 4 = 2048 waves
  const int w    = blockIdx.x * (blockDim.x >> 5) + (threadIdx.x >> 5);
  const int lane = threadIdx.x & 31;
  const int h    = w >> 8;          // head
  const int rem  = w & 255;
  const int tm   = rem >> 2;        // M-tile (output row block), 0..63
  const int tn   = rem & 3;         // N-tile (feature block),   0..3
  const int row  = tm * 16 + (lane & 15);   // A-matrix row (output index n)
  const int kofs = (lane >> 4) * 2;         // per-lane K sub-offset (A and B)
  const int z    = tn * 16 + (lane & 15);   // B-matrix column (feature z)

  // B-fragment base: all 4 planes are immediate offsets (<8MB) from one ptr
  const float* pb = ws + WS_PLANES + (h * Nn + kofs) * Dd + z;

  // Twiddle index recurrence: theta_index(r) = (row*r) mod 1032, r += 4/step.
  // step4 = (4*row) mod 1032; per step: t += step4; t -= (t>=1032)?1032:0.
  const int step4 = (row * 4) % FLEN;
  int t0 = (row * kofs) % FLEN;          // kofs in {0,2}
  int t1 = (t0 + row) % FLEN;            // index for r0+1

  v8f Gre = {}, Gim = {}, Kre = {}, Kim = {};
  const float wstep = 6.28318530717958647692f / (float)FLEN;

  for (int kk = 0; kk < Nn; kk += 4) {   // padded rows r'>=1024 are zero
    // Twiddle A-fragments: F = exp(-i*2*pi*row*r'/1032) = C - i*S
    float s0, c0, s1, c1;
    __sincosf((float)t0 * wstep, &s0, &c0);
    __sincosf((float)t1 * wstep, &s1, &c1);
    t0 += step4; if (t0 >= FLEN) t0 -= FLEN;
    t1 += step4; if (t1 >= FLEN) t1 -= FLEN;
    v2f Ac  = { c0,  c1};
    v2f As  = { s0,  s1};
    v2f Asn = {-s0, -s1};

    v2f Bgre = {pb[0],         pb[Dd]};
    v2f Bgim = {pb[PLANE],     pb[PLANE + Dd]};
    v2f Bkre = {pb[2 * PLANE], pb[2 * PLANE + Dd]};
    v2f Bkim = {pb[3 * PLANE], pb[3 * PLANE + Dd]};
    pb += 4 * Dd;

    // (C - iS)(Bre + iBim) = (C*Bre + S*Bim) + i(C*Bim - S*Bre)
    Gre = wmma4(Ac,  Bgre, Gre);  Gre = wmma4(As,  Bgim, Gre);
    Gim = wmma4(Ac,  Bgim, Gim);  Gim = wmma4(Asn, Bgre, Gim);
    Kre = wmma4(Ac,  Bkre, Kre);  Kre = wmma4(As,  Bkim, Kre);
    Kim = wmma4(Ac,  Bkim, Kim);  Kim = wmma4(Asn, Bkre, Kim);
  }

  // epilogue: out = G / K (complex), interleaved complex64
#pragma unroll
  for (int e = 0; e < 8; ++e) {
    int m = tm * 16 + e + ((lane >= 16) ? 8 : 0);   // C/D layout: M = e (+8 hi)
    float kr = Kre[e], ki = Kim[e], gr = Gre[e], gi = Gim[e];
    float inv = 1.0f / (kr * kr + ki * ki);
    float2 o;
    o.x = (gr * kr + gi * ki) * inv;
    o.y = (gi * kr - gr * ki) * inv;
    out[(h * Nn + m) * Dd + z] = o;
  }
}

extern "C" void kernel_launch(void* const* d_in, const int* in_sizes, int n_in,
                              void* d_out, int out_size, void* d_ws, size_t ws_size,
                              hipStream_t stream) {
  // setup_inputs order: q(0), k(1), v(2), Er(3), attn_mask(4), q_len(5), k_len(6)
  // q/Er/mask cancel algebraically -> only k and v are needed.
  const float* k = (const float*)d_in[1];
  const float* v = (const float*)d_in[2];
  float*  ws  = (float*)d_ws;
  float2* out = (float2*)d_out;   // complex64, interleaved (re, im)

  rpe_sumsq  <<<256,  256, 0, stream>>>(k, ws);
  rpe_finnorm<<<1,    256, 0, stream>>>(ws);
  rpe_hdft   <<<Nn,    64, 0, stream>>>(k, v, ws);
  rpe_wmma_dft<<<256, 256, 0, stream>>>(ws, out);
}